// BreedingPolicyNet_78941498901115
// MI455X (gfx1250) — compile-verified
//
#include <hip/hip_runtime.h>
#include <hip/hip_bf16.h>
#include <math.h>

// ---------------------------------------------------------------------------
// BreedingPolicyNet on MI455X (gfx1250)
//
// Roofline: per iteration the einsum must stream T (512 MB f32) once:
//   0.27 GFLOP useful / 512 MB  =>  ~0.5 FLOP/byte  =>  HBM-bound (~22us/iter
//   at 23.3 TB/s). Keep f32, coalesce all T reads, accumulate on the matrix
//   pipe with V_WMMA_F32_16X16X4_F32 (f32 in, f32 out, no precision loss).
// ---------------------------------------------------------------------------

#define N 512
#define NEG_HUGE -3.402823466e38f   // jnp.finfo(float32).min

typedef __attribute__((ext_vector_type(2))) float v2f;
typedef __attribute__((ext_vector_type(8))) float v8f;

// ---------------------------------------------------------------------------
// x_cur = max(x0, 0)
// ---------------------------------------------------------------------------
__global__ void __launch_bounds__(256)
init_x_kernel(const float* __restrict__ x0, float* __restrict__ xc) {
    int k = blockIdx.x * 256 + threadIdx.x;
    if (k < N) xc[k] = fmaxf(x0[k], 0.0f);
}

// ---------------------------------------------------------------------------
// w[a,b] = x[a] * Qi[a,b]   (masked row softmax; Qi row zero if no allowed)
// one block per row a; 256 threads handle 2 columns each
// ---------------------------------------------------------------------------
__global__ void __launch_bounds__(256)
softmax_w_kernel(const float* __restrict__ x,
                 const float* __restrict__ logits,
                 const int* __restrict__ tgt,
                 float* __restrict__ wbuf) {
    __shared__ float s_red[256];
    __shared__ int   s_cnt[256];

    const int a      = blockIdx.x;
    const int t      = threadIdx.x;
    const int target = tgt[0];

    const float xa = x[a];
    const bool  pa = xa > 0.0f;

    const int b0 = t;
    const int b1 = t + 256;
    const bool al0 = pa && (x[b0] > 0.0f) && (b0 != target);
    const bool al1 = pa && (x[b1] > 0.0f) && (b1 != target);
    const float l0 = al0 ? logits[a * N + b0] : NEG_HUGE;
    const float l1 = al1 ? logits[a * N + b1] : NEG_HUGE;

    // row max
    s_red[t] = fmaxf(l0, l1);
    __syncthreads();
    for (int s = 128; s > 0; s >>= 1) {
        if (t < s) s_red[t] = fmaxf(s_red[t], s_red[t + s]);
        __syncthreads();
    }
    const float rowmax = s_red[0];
    __syncthreads();

    // exp + row sum + allowed count
    const float e0 = expf(l0 - rowmax);   // masked entries underflow to 0
    const float e1 = expf(l1 - rowmax);
    s_red[t] = e0 + e1;
    s_cnt[t] = (int)al0 + (int)al1;
    __syncthreads();
    for (int s = 128; s > 0; s >>= 1) {
        if (t < s) { s_red[t] += s_red[t + s]; s_cnt[t] += s_cnt[t + s]; }
        __syncthreads();
    }
    const float rowsum = s_red[0];
    const int   cnt    = s_cnt[0];

    const float scale = (cnt > 0) ? (xa / rowsum) : 0.0f;
    wbuf[a * N + b0] = e0 * scale;
    wbuf[a * N + b1] = e1 * scale;
}

// ---------------------------------------------------------------------------
// partial[bx*512 + k] = sum over 1024 rows of w[r] * T[r][k]
//
// T viewed as [262144, 512] f32 (row = a*512+b, col = k, rows contiguous).
// grid = (256 row-chunks, 4 tile-groups); block = 256 threads = 8 waves.
// Each wave owns one 16-wide k-tile (tile = by*8 + wave) and marches 256
// groups of 4 rows, issuing one V_WMMA_F32_16X16X4_F32 per group:
//   A (16x4): row m = {w[r+2h], w[r+2h+1]}  (identical rows -> GEMV)
//   B (4x16): lane n, half h holds T[r+2h][kbase+n], T[r+2h+1][kbase+n]
// Lanes 0-15 read 64B contiguous per row -> fully coalesced; every T byte is
// touched exactly once per iteration.
// ---------------------------------------------------------------------------
__global__ void __launch_bounds__(256)
gemv_wmma_kernel(const float* __restrict__ T,
                 const float* __restrict__ wbuf,
                 float* __restrict__ partial) {
    const int lane = threadIdx.x & 31;
    const int wave = threadIdx.x >> 5;
    const int half = lane >> 4;
    const int n    = lane & 15;

    const int tile  = blockIdx.y * 8 + wave;   // 0..31
    const int kbase = tile * 16;
    const int rbase = blockIdx.x * 1024;

    const float* __restrict__ Tcol = T + (size_t)rbase * N + kbase + n;
    const float* __restrict__ wp   = wbuf + rbase;

    v8f c = {};
    for (int g = 0; g < 256; ++g) {
        const int r2 = g * 4 + 2 * half;       // even -> 8B aligned w load
        v2f a = *(const v2f*)(wp + r2);        // {w[r2], w[r2+1]}
        v2f b;
        b.x = Tcol[(size_t)r2 * N];
        b.y = Tcol[(size_t)(r2 + 1) * N];
        c = __builtin_amdgcn_wmma_f32_16x16x4_f32(
                /*neg_a=*/false, a, /*neg_b=*/false, b,
                /*c_mod=*/(short)0, c, /*reuse_a=*/false, /*reuse_b=*/false);
    }

    // All 16 C rows identical; row M=0 lives in c[0] of lanes 0..15.
    if (lane < 16) partial[blockIdx.x * N + kbase + n] = c[0];
}

// ---------------------------------------------------------------------------
// x_out[k] = max(x_in[k] + sum_bx partial[bx*512 + k], 0)
// ---------------------------------------------------------------------------
__global__ void __launch_bounds__(256)
reduce_update_kernel(const float* __restrict__ partial,
                     const float* __restrict__ x_in,
                     float* __restrict__ x_out) {
    const int k = blockIdx.x * 256 + threadIdx.x;
    if (k >= N) return;
    float s = 0.0f;
    for (int bx = 0; bx < 256; ++bx) s += partial[bx * N + k];
    x_out[k] = fmaxf(x_in[k] + s, 0.0f);
}

// ---------------------------------------------------------------------------
extern "C" void kernel_launch(void* const* d_in, const int* in_sizes, int n_in,
                              void* d_out, int out_size, void* d_ws, size_t ws_size,
                              hipStream_t stream) {
    const float* x0     = (const float*)d_in[0];
    const float* logits = (const float*)d_in[1];
    const float* T      = (const float*)d_in[2];
    const int*   tgt    = (const int*)d_in[3];
    float*       out    = (float*)d_out;

    float* ws      = (float*)d_ws;
    float* wbuf    = ws;                         // 262144 floats (1 MB)
    float* partial = ws + 262144;                // 131072 floats (512 KB)
    float* xc      = ws + 262144 + 131072;       // 512 floats

    (void)in_sizes; (void)n_in; (void)out_size; (void)ws_size;

    init_x_kernel<<<2, 256, 0, stream>>>(x0, xc);

    for (int it = 0; it < 8; ++it) {
        softmax_w_kernel<<<N, 256, 0, stream>>>(xc, logits, tgt, wbuf);
        gemv_wmma_kernel<<<dim3(256, 4), 256, 0, stream>>>(T, wbuf, partial);
        float* dst = (it == 7) ? out : xc;
        reduce_update_kernel<<<2, 256, 0, stream>>>(partial, xc, dst);
    }
}